// GraphAttentionLayer_28518582846264
// MI455X (gfx1250) — compile-verified
//
#include <hip/hip_runtime.h>
#include <math.h>

// ---------------- problem constants ----------------
#define N_NODES   100000
#define N_EDGES   1600000
#define HIDDEN    128
#define HEADS     8
#define HEAD_DIM  16
#define ALPHA     0.2f
#define LN_EPS    1e-5f

typedef __attribute__((ext_vector_type(2))) float v2f;
typedef __attribute__((ext_vector_type(8))) float v8f;

// ---------------- init: zero accumulator, -inf segment-max, zero denom ----------------
__global__ void init_kernel(float* __restrict__ out, float* __restrict__ m,
                            float* __restrict__ denom) {
    long long i = (long long)blockIdx.x * blockDim.x + threadIdx.x;
    if (i < (long long)N_NODES * HIDDEN) out[i] = 0.0f;
    if (i < (long long)N_NODES * HEADS) {
        m[i]     = -INFINITY;
        denom[i] = 0.0f;
    }
}

// ---------------- x_heads = X @ W  via V_WMMA_F32_16X16X4_F32 ----------------
// Grid: 6250 blocks (row tiles of 16). Block: 256 threads = 8 waves; wave w == head w.
// xh layout: (N_NODES, HIDDEN) with column h*16+j == x_heads[n][h][j].
__global__ __launch_bounds__(256)
void xheads_gemm(const float* __restrict__ X, const float* __restrict__ W,
                 float* __restrict__ xh) {
    const int wave = threadIdx.x >> 5;      // head index 0..7
    const int lane = threadIdx.x & 31;
    const int mlo  = lane & 15;
    const bool hi  = lane >= 16;
    const int rowTile = blockIdx.x * 16;
    const int m = rowTile + mlo;            // A row for this lane
    const float* __restrict__ Wh = W + wave * (HIDDEN * HEAD_DIM); // (128 x 16) row-major

    v8f c = {};
    #pragma unroll 4
    for (int k = 0; k < HIDDEN; k += 4) {
        const int ka = k + (hi ? 2 : 0);    // lanes 16-31 carry K+2, K+3
        v2f a, b;
        a.x = X[(long long)m * HIDDEN + ka];
        a.y = X[(long long)m * HIDDEN + ka + 1];
        b.x = Wh[ka * HEAD_DIM + mlo];          // B[k][n]
        b.y = Wh[(ka + 1) * HEAD_DIM + mlo];    // B[k+1][n]
        c = __builtin_amdgcn_wmma_f32_16x16x4_f32(
                /*neg_a=*/false, a, /*neg_b=*/false, b,
                /*c_mod=*/(short)0, c, /*reuse_a=*/false, /*reuse_b=*/false);
    }
    // D layout: VGPR r -> M = r (lanes 0-15) / r+8 (lanes 16-31), N = lane%16
    const int col = wave * HEAD_DIM + mlo;
    #pragma unroll
    for (int r = 0; r < 8; ++r) {
        const int row = rowTile + r + (hi ? 8 : 0);
        xh[(long long)row * HIDDEN + col] = c[r];
    }
}

// ---------------- s_i[n][h], s_j[n][h] ----------------
__global__ void si_sj_kernel(const float* __restrict__ xh, const float* __restrict__ a,
                             float* __restrict__ s_i, float* __restrict__ s_j) {
    const int t = blockIdx.x * blockDim.x + threadIdx.x;   // t = n*8 + h
    if (t >= N_NODES * HEADS) return;
    const int n = t >> 3, h = t & 7;
    const float* __restrict__ v  = xh + (long long)n * HIDDEN + h * HEAD_DIM;
    const float* __restrict__ ai = a + h * (2 * HEAD_DIM);
    const float* __restrict__ aj = ai + HEAD_DIM;
    float si = 0.0f, sj = 0.0f;
    #pragma unroll
    for (int j = 0; j < HEAD_DIM; ++j) {
        const float xv = v[j];
        si = fmaf(xv, ai[j], si);
        sj = fmaf(xv, aj[j], sj);
    }
    s_i[t] = si;
    s_j[t] = sj;
}

// sign-split float atomic max (correct for mixed signs when init is -inf bits)
__device__ __forceinline__ void atomicMaxF(float* addr, float v) {
    if (v >= 0.0f) atomicMax((int*)addr, __float_as_int(v));
    else           atomicMin((unsigned int*)addr, __float_as_uint(v));
}

// ---------------- edge pass 1: segment max of leaky-relu logits ----------------
__global__ void edge_max_kernel(const int* __restrict__ rowI, const int* __restrict__ colI,
                                const float* __restrict__ s_i, const float* __restrict__ s_j,
                                float* __restrict__ m) {
    const int e = blockIdx.x * blockDim.x + threadIdx.x;
    if (e >= N_EDGES) return;
    const int r = rowI[e], c = colI[e];
    #pragma unroll
    for (int h = 0; h < HEADS; ++h) {
        float ev = s_i[r * HEADS + h] + s_j[c * HEADS + h];
        ev = ev >= 0.0f ? ev : ALPHA * ev;
        atomicMaxF(&m[r * HEADS + h], ev);
    }
}

// ---------------- edge pass 2: denom = segment sum of exp(e - m) ----------------
__global__ void edge_denom_kernel(const int* __restrict__ rowI, const int* __restrict__ colI,
                                  const float* __restrict__ s_i, const float* __restrict__ s_j,
                                  const float* __restrict__ m, float* __restrict__ denom) {
    const int e = blockIdx.x * blockDim.x + threadIdx.x;
    if (e >= N_EDGES) return;
    const int r = rowI[e], c = colI[e];
    #pragma unroll
    for (int h = 0; h < HEADS; ++h) {
        float ev = s_i[r * HEADS + h] + s_j[c * HEADS + h];
        ev = ev >= 0.0f ? ev : ALPHA * ev;
        atomicAdd(&denom[r * HEADS + h], expf(ev - m[r * HEADS + h]));
    }
}

// ---------------- edge pass 3: out[col] += attn * x_heads[row]  (wave per edge) ----------------
__global__ __launch_bounds__(256)
void edge_scatter_kernel(const int* __restrict__ rowI, const int* __restrict__ colI,
                         const float* __restrict__ s_i, const float* __restrict__ s_j,
                         const float* __restrict__ m, const float* __restrict__ denom,
                         const float* __restrict__ xh, float* __restrict__ out) {
    const long long gid = (long long)blockIdx.x * blockDim.x + threadIdx.x;
    const long long e = gid >> 5;            // one wave (32 lanes) per edge
    const int lane = (int)(gid & 31);        // lane handles channels [lane*4, lane*4+4)
    if (e >= N_EDGES) return;
    const int r = rowI[e], c = colI[e];
    const int h = lane >> 2;                 // head = channel/16
    float ev = s_i[r * HEADS + h] + s_j[c * HEADS + h];
    ev = ev >= 0.0f ? ev : ALPHA * ev;
    const float attn = expf(ev - m[r * HEADS + h]) / denom[r * HEADS + h];
    const float4 w = *reinterpret_cast<const float4*>(xh + (long long)r * HIDDEN + lane * 4);
    float* o = out + (long long)c * HIDDEN + lane * 4;
    atomicAdd(o + 0, w.x * attn);
    atomicAdd(o + 1, w.y * attn);
    atomicAdd(o + 2, w.z * attn);
    atomicAdd(o + 3, w.w * attn);
}

// ---------------- residual + LayerNorm (in place on d_out) ----------------
__global__ __launch_bounds__(128)
void ln_kernel(const float* __restrict__ x, const float* __restrict__ gamma,
               const float* __restrict__ beta, float* __restrict__ out) {
    const int n = blockIdx.x;
    const int t = threadIdx.x;               // 128 threads = 4 waves
    float v = out[(long long)n * HIDDEN + t] + x[(long long)n * HIDDEN + t];

    __shared__ float red[4];
    float s = v;
    #pragma unroll
    for (int off = 16; off > 0; off >>= 1) s += __shfl_down(s, off);
    if ((t & 31) == 0) red[t >> 5] = s;
    __syncthreads();
    const float mu = (red[0] + red[1] + red[2] + red[3]) * (1.0f / HIDDEN);
    __syncthreads();

    const float d = v - mu;
    float s2 = d * d;
    #pragma unroll
    for (int off = 16; off > 0; off >>= 1) s2 += __shfl_down(s2, off);
    if ((t & 31) == 0) red[t >> 5] = s2;
    __syncthreads();
    const float var = (red[0] + red[1] + red[2] + red[3]) * (1.0f / HIDDEN);

    out[(long long)n * HIDDEN + t] = d * rsqrtf(var + LN_EPS) * gamma[t] + beta[t];
}

// ---------------- launch ----------------
extern "C" void kernel_launch(void* const* d_in, const int* in_sizes, int n_in,
                              void* d_out, int out_size, void* d_ws, size_t ws_size,
                              hipStream_t stream) {
    const float* x     = (const float*)d_in[0];
    const int*   ei    = (const int*)d_in[1];     // (2, N_EDGES): row then col
    const float* W     = (const float*)d_in[2];   // (HEADS, HIDDEN, HEAD_DIM)
    const float* avec  = (const float*)d_in[3];   // (HEADS, 2*HEAD_DIM, 1)
    const float* gamma = (const float*)d_in[4];
    const float* beta  = (const float*)d_in[5];
    float*       out   = (float*)d_out;

    const int* rowI = ei;
    const int* colI = ei + N_EDGES;

    // workspace layout (floats):
    float* ws   = (float*)d_ws;
    float* xh   = ws;                                  // N_NODES*HIDDEN = 12.8M
    float* s_i  = xh  + (long long)N_NODES * HIDDEN;   // N_NODES*HEADS
    float* s_j  = s_i + (long long)N_NODES * HEADS;
    float* mbuf = s_j + (long long)N_NODES * HEADS;
    float* den  = mbuf + (long long)N_NODES * HEADS;

    // 1) init accumulator (d_out), segment-max, denom
    {
        const long long tot = (long long)N_NODES * HIDDEN;
        const int blocks = (int)((tot + 255) / 256);
        init_kernel<<<blocks, 256, 0, stream>>>(out, mbuf, den);
    }
    // 2) x_heads GEMM via WMMA (6250 row tiles, wave-per-head)
    xheads_gemm<<<N_NODES / 16, 256, 0, stream>>>(x, W, xh);
    // 3) attention logits per node/head
    si_sj_kernel<<<(N_NODES * HEADS + 255) / 256, 256, 0, stream>>>(xh, avec, s_i, s_j);
    // 4) edge passes
    edge_max_kernel<<<(N_EDGES + 255) / 256, 256, 0, stream>>>(rowI, colI, s_i, s_j, mbuf);
    edge_denom_kernel<<<(N_EDGES + 255) / 256, 256, 0, stream>>>(rowI, colI, s_i, s_j, mbuf, den);
    {
        const long long tot = (long long)N_EDGES * 32;
        const int blocks = (int)((tot + 255) / 256);
        edge_scatter_kernel<<<blocks, 256, 0, stream>>>(rowI, colI, s_i, s_j, mbuf, den, xh, out);
    }
    // 5) residual + LayerNorm in place
    ln_kernel<<<N_NODES, 128, 0, stream>>>(x, gamma, beta, out);
}